// GuidedFilter_11304353923476
// MI455X (gfx1250) — compile-verified
//
#include <hip/hip_runtime.h>

// Guided filter (r=15) for B=8, C=3, H=W=512 f32, on gfx1250.
// Box filters implemented as block-banded matmuls via V_WMMA_F32_16X16X4_F32:
// a 1-D box sum is a multiply by a banded 0/1 matrix; per 16x16 output tile
// only K in [tile-16, tile+32) contributes -> 12 WMMA(K=4) ops per tile, in
// full f32 (same arithmetic class as the reference). Per-batch pipeline keeps
// all intermediates (~36 MB) resident in the 192 MB L2, so HBM traffic is
// just the compulsory input read + output write (~75 MB @ 23.3 TB/s).
// Pure-copy LDS staging uses GLOBAL_LOAD_ASYNC_TO_LDS_B128 (ASYNCcnt path).

typedef float v2f  __attribute__((ext_vector_type(2)));
typedef float v8f  __attribute__((ext_vector_type(8)));
typedef float fv4  __attribute__((ext_vector_type(4)));
typedef int   i4v  __attribute__((ext_vector_type(4)));

#define H_     512
#define W_     512
#define NPIX   (H_ * W_)          // 262144 = 1 << 18
#define INV_K2 (1.0f / 961.0f)    // 1/(31*31)
#define PADH   180                // H-pass LDS row stride (16B aligned, conflict-free)
#define PADV   136                // V-pass LDS row stride (16B aligned, conflict-free)

#if defined(__has_builtin)
#  if __has_builtin(__builtin_amdgcn_global_load_async_to_lds_b128) && \
      __has_builtin(__builtin_amdgcn_s_wait_asynccnt)
#    define USE_ASYNC_LDS 1
#  endif
#endif
#ifndef USE_ASYNC_LDS
#  define USE_ASYNC_LDS 0
#endif

#if USE_ASYNC_LDS
__device__ __forceinline__ void async_copy_b128(const float* gsrc, float* ldst) {
  // Signature: (AS1 int4* src, AS3 int4* dst, imm offset, imm cpol)
  __builtin_amdgcn_global_load_async_to_lds_b128(
      (__attribute__((address_space(1))) i4v*)gsrc,
      (__attribute__((address_space(3))) i4v*)ldst, 0, 0);
}
#endif

__device__ __forceinline__ v8f wmma_f32_16x16x4(v2f a, v2f b, v8f c) {
  // 8 args: (neg_a, A, neg_b, B, c_mod, C, reuse_a, reuse_b)
  return __builtin_amdgcn_wmma_f32_16x16x4_f32(false, a, false, b, (short)0, c,
                                               false, false);
}

// ---------------------------------------------------------------- gray -----
__global__ void __launch_bounds__(256)
k_gray(const float* __restrict__ guide, float* __restrict__ gray) {
  int idx  = blockIdx.x * 256 + threadIdx.x;   // over 8*NPIX/4 chunks
  int b    = idx >> 16;                        // NPIX/4 = 65536
  int pix  = (idx & 0xFFFF) * 4;
  const float* gb = guide + (size_t)b * 3 * NPIX;
  fv4 r  = *(const fv4*)(gb + pix);
  fv4 g  = *(const fv4*)(gb + NPIX + pix);
  fv4 bl = *(const fv4*)(gb + 2 * NPIX + pix);
  *(fv4*)(gray + (size_t)b * NPIX + pix) = 0.299f * r + 0.587f * g + 0.114f * bl;
}

// -------------------------------------------------- horizontal box pass ----
// MODE 0: produce plane element on the fly: type 0: g, 1: g*g, 2..4: p[c],
//         5..7: g*p[c].   MODE 1: direct copy from src planes (async path).
// Block: 8 wave32; rows [R0,R0+16) x cols [C0,C0+128). Each wave owns a 16x16
// tile of S = X * Aband: X from LDS as the A-operand, the 0/1 band generated
// in registers as the B-operand, 12 x WMMA_F32_16X16X4_F32.
template <int MODE>
__global__ void __launch_bounds__(256)
k_boxh(const float* __restrict__ g, const float* __restrict__ p,
       float* __restrict__ dst) {
  __shared__ float tile[16 * PADH];            // 176 data cols + pad
  const int type = blockIdx.z;
  const int R0   = blockIdx.y * 16;
  const int C0   = blockIdx.x * 128;
  const int tid  = threadIdx.y * 32 + threadIdx.x;

  // Stage rows [R0,R0+16) x global cols [C0-16, C0+160) in 16B chunks.
  for (int i = tid; i < 16 * 44; i += 256) {
    int r  = i / 44;
    int c4 = (i - r * 44) * 4;
    int gc = C0 - 16 + c4;                     // chunk fully in or out of image
    float* lp = &tile[r * PADH + c4];
    if ((unsigned)gc < (unsigned)W_) {
      size_t off = (size_t)(R0 + r) * W_ + gc;
#if USE_ASYNC_LDS
      if (MODE == 1) {
        async_copy_b128(g + (size_t)type * NPIX + off, lp);
        continue;
      }
#endif
      fv4 v;
      if (MODE == 1) {
        v = *(const fv4*)(g + (size_t)type * NPIX + off);
      } else {
        if (type == 0)      { v = *(const fv4*)(g + off); }
        else if (type == 1) { fv4 gv = *(const fv4*)(g + off); v = gv * gv; }
        else if (type <= 4) { v = *(const fv4*)(p + (size_t)(type - 2) * NPIX + off); }
        else                { fv4 gv = *(const fv4*)(g + off);
                              v = gv * *(const fv4*)(p + (size_t)(type - 5) * NPIX + off); }
      }
      *(fv4*)lp = v;
    } else {
      fv4 z = {};
      *(fv4*)lp = z;
    }
  }
#if USE_ASYNC_LDS
  if (MODE == 1) __builtin_amdgcn_s_wait_asynccnt(0);
#endif
  __syncthreads();

  const int lane = threadIdx.x;
  const int w    = threadIdx.y;       // wave id -> 16-col tile within 128
  const int m    = lane & 15;         // A-matrix row / D column index
  const int half = lane >> 4;
  const int ka   = half * 2;          // K base for this half-wave (0 or 2)
  const int t    = ka - 1 - m;        // dj - m + 15 = 4s + t

  v8f acc = {};
#pragma unroll
  for (int s = 0; s < 12; ++s) {
    int lj = w * 16 + 4 * s;          // LDS col of K base (LDS col0 = C0-16)
    v2f a, b;
    a.x = tile[m * PADH + lj + ka];
    a.y = tile[m * PADH + lj + ka + 1];
    unsigned u = (unsigned)(4 * s + t);
    b.x = (u     <= 30u) ? 1.0f : 0.0f;
    b.y = (u + 1 <= 30u) ? 1.0f : 0.0f;
    acc = wmma_f32_16x16x4(a, b, acc);
  }

  float* out = dst + (size_t)type * NPIX;
  const int Ct0 = C0 + w * 16;
#pragma unroll
  for (int e = 0; e < 8; ++e)
    out[(size_t)(R0 + e + 8 * half) * W_ + Ct0 + m] = acc[e];
}

// ---------------------------------------------------- vertical box pass ----
// Block covers rows [R0,R0+16) x cols [C0,C0+128); stages 48 input rows in
// LDS (stride 136 -> the +2-row half-wave K offset lands on a disjoint
// 16-bank window, conflict-free). S = Aband * X: band in registers as the
// A-operand, data from LDS as the B-operand. Applies 1/961 on store.
__global__ void __launch_bounds__(256)
k_boxv(const float* __restrict__ src, float* __restrict__ dst, float scale) {
  __shared__ float tile[48 * PADV];
  const int type = blockIdx.z;
  const int R0   = blockIdx.y * 16;
  const int C0   = blockIdx.x * 128;
  const int tid  = threadIdx.y * 32 + threadIdx.x;
  const float* sp = src + (size_t)type * NPIX;

  for (int i = tid; i < 48 * 32; i += 256) {   // 16B chunks
    int r  = i >> 5;
    int c4 = (i & 31) * 4;
    int gr = R0 - 16 + r;
    float* lp = &tile[r * PADV + c4];
    if ((unsigned)gr < (unsigned)H_) {
      const float* gp = sp + (size_t)gr * W_ + C0 + c4;
#if USE_ASYNC_LDS
      async_copy_b128(gp, lp);
#else
      *(fv4*)lp = *(const fv4*)gp;
#endif
    } else {
      fv4 z = {};
      *(fv4*)lp = z;
    }
  }
#if USE_ASYNC_LDS
  __builtin_amdgcn_s_wait_asynccnt(0);
#endif
  __syncthreads();

  const int lane = threadIdx.x;
  const int w    = threadIdx.y;
  const int n    = lane & 15;         // B-matrix col / A-matrix row index
  const int half = lane >> 4;
  const int kb   = half * 2;
  const int t    = kb - 1 - n;        // dj - n + 15 = 4s + t

  v8f acc = {};
#pragma unroll
  for (int s = 0; s < 12; ++s) {
    int lr = 4 * s;                   // LDS row of K base (LDS row0 = R0-16)
    v2f a, b;
    b.x = tile[(lr + kb)     * PADV + w * 16 + n];
    b.y = tile[(lr + kb + 1) * PADV + w * 16 + n];
    unsigned u = (unsigned)(4 * s + t);
    a.x = (u     <= 30u) ? 1.0f : 0.0f;
    a.y = (u + 1 <= 30u) ? 1.0f : 0.0f;
    acc = wmma_f32_16x16x4(a, b, acc);
  }

  float* out = dst + (size_t)type * NPIX;
#pragma unroll
  for (int e = 0; e < 8; ++e)
    out[(size_t)(R0 + e + 8 * half) * W_ + C0 + w * 16 + n] = acc[e] * scale;
}

// --------------------------------------------------------- a,b per pixel ---
__global__ void __launch_bounds__(256)
k_ab(const float* __restrict__ means, float* __restrict__ ab) {
  int idx = blockIdx.x * 256 + threadIdx.x;    // over 3*NPIX/4 chunks
  int c   = idx >> 16;
  int pix = (idx & 0xFFFF) * 4;
  fv4 mI  = *(const fv4*)(means + pix);
  fv4 cI  = *(const fv4*)(means + NPIX + pix);
  fv4 mp  = *(const fv4*)(means + (size_t)(2 + c) * NPIX + pix);
  fv4 cIp = *(const fv4*)(means + (size_t)(5 + c) * NPIX + pix);
  fv4 varI = cI - mI * mI;
  fv4 cov  = cIp - mI * mp;
  fv4 a    = cov / (varI + 1e-6f);
  *(fv4*)(ab + (size_t)c * NPIX + pix)       = a;
  *(fv4*)(ab + (size_t)(3 + c) * NPIX + pix) = mp - a * mI;
}

// ------------------------------------------------------------- output ------
__global__ void __launch_bounds__(256)
k_final(const float* __restrict__ mab, const float* __restrict__ gray,
        float* __restrict__ out) {
  int idx = blockIdx.x * 256 + threadIdx.x;    // over 3*NPIX/4 chunks
  int c   = idx >> 16;
  int pix = (idx & 0xFFFF) * 4;
  fv4 gv = *(const fv4*)(gray + pix);
  fv4 ma = *(const fv4*)(mab + (size_t)c * NPIX + pix);
  fv4 mb = *(const fv4*)(mab + (size_t)(3 + c) * NPIX + pix);
  *(fv4*)(out + idx * 4) = ma * gv + mb;
}

// ---------------------------------------------------------------------------
extern "C" void kernel_launch(void* const* d_in, const int* in_sizes, int n_in,
                              void* d_out, int out_size, void* d_ws, size_t ws_size,
                              hipStream_t stream) {
  const float* guide = (const float*)d_in[0];   // [8,3,512,512]
  const float* input = (const float*)d_in[1];   // [8,3,512,512]
  float* out = (float*)d_out;                   // [8,3,512,512]
  float* ws  = (float*)d_ws;

  // Workspace layout (planes of NPIX floats): 36 planes = 36 MB total.
  float* GRAY  = ws;                            // 8 planes (all batches)
  float* HTMP  = GRAY  + (size_t)8 * NPIX;      // 8 planes (per-batch scratch)
  float* MEANS = HTMP  + (size_t)8 * NPIX;      // 8 planes
  float* AB    = MEANS + (size_t)8 * NPIX;      // 6 planes
  float* MAB   = AB    + (size_t)6 * NPIX;      // 6 planes

  k_gray<<<(8 * NPIX / 4) / 256, 256, 0, stream>>>(guide, GRAY);

  dim3 blk(32, 8);                              // 8 wave32 per block
  dim3 grd8(W_ / 128, H_ / 16, 8);
  dim3 grd6(W_ / 128, H_ / 16, 6);

  for (int b = 0; b < 8; ++b) {
    const float* gb = GRAY  + (size_t)b * NPIX;
    const float* pb = input + (size_t)b * 3 * NPIX;
    // box(gray, gray^2, p0..2, g*p0..2)
    k_boxh<0><<<grd8, blk, 0, stream>>>(gb, pb, HTMP);
    k_boxv<<<grd8, blk, 0, stream>>>(HTMP, MEANS, INV_K2);
    // a,b
    k_ab<<<(3 * NPIX / 4) / 256, 256, 0, stream>>>(MEANS, AB);
    // box(a0..2, b0..2)
    k_boxh<1><<<grd6, blk, 0, stream>>>(AB, nullptr, HTMP);
    k_boxv<<<grd6, blk, 0, stream>>>(HTMP, MAB, INV_K2);
    // out = mean_a * gray + mean_b
    k_final<<<(3 * NPIX / 4) / 256, 256, 0, stream>>>(MAB, gb, out + (size_t)b * 3 * NPIX);
  }
}